// GCNNodeClassifier_62371515072942
// MI455X (gfx1250) — compile-verified
//
#include <hip/hip_runtime.h>
#include <cstdint>

typedef float v2f __attribute__((ext_vector_type(2)));
typedef float v8f __attribute__((ext_vector_type(8)));

#define COLS 64           // all hidden GEMMs have 64 output columns

// ---------------- degree / normalization ----------------
__global__ void k_fill1(float* p, int n) {
    int i = blockIdx.x * blockDim.x + threadIdx.x;
    if (i < n) p[i] = 1.0f;                    // self-loop contributes 1 to every degree
}

__global__ void k_deg(const int* __restrict__ dst, float* __restrict__ deg, int E) {
    int e = blockIdx.x * blockDim.x + threadIdx.x;
    if (e < E) unsafeAtomicAdd(&deg[dst[e]], 1.0f);   // global_atomic_add_f32
}

__global__ void k_rsqrt(float* p, int n) {
    int i = blockIdx.x * blockDim.x + threadIdx.x;
    if (i < n) p[i] = rsqrtf(p[i]);            // deg >= 1 always (self-loops)
}

// ---------------- fp32 WMMA GEMM:  Y[n x 64] = X[n x K] @ W[K x 64] (+bias)(+relu) ----
// One wave -> 32 rows x 64 cols (two M-tiles, four N-tiles, eight 16x16 f32 accum).
// Block = 256 threads = 8 waves = 256 rows.
// W is staged TRANSPOSED in LDS ([64][K+4]) via async global->LDS copies so the two
// B elements {W[k][c], W[k+1][c]} are contiguous -> single conflict-free ds_load_b64.
__global__ void k_gemm_wmma(const float* __restrict__ X, const float* __restrict__ W,
                            const float* __restrict__ bias, float* __restrict__ Y,
                            int n, int K, int do_relu) {
    extern __shared__ float ldsw[];            // 64 * (K+4) floats, transposed W
    const int tid   = threadIdx.x;
    const int pitch = K + 4;

    // ---- async stage W (transposed) into LDS: GLOBAL_LOAD_ASYNC_TO_LDS_B32 ----
    for (int i = tid; i < K * COLS; i += 256) {
        const int k = i >> 6;                  // row of W
        const int c = i & 63;                  // col of W
        const uint32_t ldst = (uint32_t)(size_t)(&ldsw[c * pitch + k]); // LDS byte offset
        asm volatile("global_load_async_to_lds_b32 %0, %1, off"
                     :: "v"(ldst), "v"(W + i) : "memory");
    }
    asm volatile("s_wait_asynccnt 0" ::: "memory");
    __syncthreads();

    const int lane = tid & 31;
    const int wave = tid >> 5;
    const int rowBase = blockIdx.x * 256 + wave * 32;
    if (rowBase >= n) return;                  // whole 32-row slab out of range

    // A-matrix (16x4 f32) layout: lanes 0-15 hold M=lane with K{0,1} in the VGPR pair;
    // lanes 16-31 hold M=lane-16 with K{2,3}.
    const int kHalf = (lane >> 4) * 2;         // 0 for lanes 0-15, 2 for lanes 16-31
    const int col   = lane & 15;
    int r0 = rowBase + col;        if (r0 >= n) r0 = n - 1;   // clamp: EXEC must stay all-1s
    int r1 = rowBase + 16 + col;   if (r1 >= n) r1 = n - 1;
    const float* __restrict__ x0 = X + (size_t)r0 * K;
    const float* __restrict__ x1 = X + (size_t)r1 * K;

    v8f acc[2][4];
#pragma unroll
    for (int t = 0; t < 2; ++t)
#pragma unroll
        for (int nt = 0; nt < 4; ++nt) acc[t][nt] = v8f{};

    for (int k4 = 0; k4 < K; k4 += 4) {
        const int kb = k4 + kHalf;
        __builtin_prefetch(x0 + k4 + 64, 0, 3);            // global_prefetch_b8
        v2f a0 = *(const v2f*)(x0 + kb);
        v2f a1 = *(const v2f*)(x1 + kb);
#pragma unroll
        for (int nt = 0; nt < 4; ++nt) {
            // B-matrix (4x16 f32): lanes 0-15 rows {0,1}, lanes 16-31 rows {2,3};
            // transposed LDS makes the pair one aligned b64 load.
            v2f b = *(const v2f*)(&ldsw[(nt * 16 + col) * pitch + kb]);
            acc[0][nt] = __builtin_amdgcn_wmma_f32_16x16x4_f32(
                false, a0, false, b, (short)0, acc[0][nt], false, false);
            acc[1][nt] = __builtin_amdgcn_wmma_f32_16x16x4_f32(
                false, a1, false, b, (short)0, acc[1][nt], false, false);
        }
    }

    // C/D layout: VGPR r -> M = r (lanes 0-15) / M = r+8 (lanes 16-31), N = lane&15
    const int mOff = (lane >> 4) * 8;
#pragma unroll
    for (int t = 0; t < 2; ++t) {
#pragma unroll
        for (int r = 0; r < 8; ++r) {
            const int row = rowBase + t * 16 + mOff + r;
            if (row < n) {
                float* yr = Y + (size_t)row * COLS;
#pragma unroll
                for (int nt = 0; nt < 4; ++nt) {
                    float v = acc[t][nt][r];
                    const int gc = nt * 16 + col;
                    if (bias)    v += bias[gc];
                    if (do_relu) v = fmaxf(v, 0.0f);
                    yr[gc] = v;
                }
            }
        }
    }
}

// ---------------- aggregation ----------------
__global__ void k_agg_init(const float* __restrict__ H, const float* __restrict__ dinv,
                           float* __restrict__ agg, int n) {
    int i = blockIdx.x * blockDim.x + threadIdx.x;
    if (i < n * COLS) {
        float di = dinv[i >> 6];
        agg[i] = H[i] * di * di;               // self-loop term
    }
}

__global__ void k_scatter(const int* __restrict__ src, const int* __restrict__ dst,
                          const float* __restrict__ dinv, const float* __restrict__ H,
                          float* __restrict__ agg, int E) {
    int t = blockIdx.x * blockDim.x + threadIdx.x;
    int e = t >> 5;
    if (e >= E) return;
    int lane = t & 31;
    int s = src[e], d = dst[e];
    float w = dinv[s] * dinv[d];
    const float2 h2 = *(const float2*)(H + (size_t)s * COLS + lane * 2);
    unsafeAtomicAdd(&agg[(size_t)d * COLS + lane * 2 + 0], h2.x * w);
    unsafeAtomicAdd(&agg[(size_t)d * COLS + lane * 2 + 1], h2.y * w);
}

__global__ void k_finalize(float* __restrict__ agg, const float* __restrict__ bias, int n) {
    int i = blockIdx.x * blockDim.x + threadIdx.x;
    if (i < n * COLS) agg[i] = fmaxf(agg[i] + bias[i & 63], 0.0f);
}

// ---------------- classifier head: logits = z @ Wc + bc  (64 -> 6) ----------------
__global__ void k_logits(const float* __restrict__ z, const float* __restrict__ Wc,
                         const float* __restrict__ bc, float* __restrict__ out, int n) {
    int node = blockIdx.x * blockDim.x + threadIdx.x;
    if (node >= n) return;
    float acc[6];
#pragma unroll
    for (int c = 0; c < 6; ++c) acc[c] = bc[c];
    const float* zr = z + (size_t)node * COLS;
    for (int k = 0; k < COLS; ++k) {
        float zv = zr[k];
#pragma unroll
        for (int c = 0; c < 6; ++c) acc[c] += zv * Wc[k * 6 + c];
    }
#pragma unroll
    for (int c = 0; c < 6; ++c) out[(size_t)node * 6 + c] = acc[c];
}

// ---------------- host launcher ----------------
extern "C" void kernel_launch(void* const* d_in, const int* in_sizes, int n_in,
                              void* d_out, int out_size, void* d_ws, size_t ws_size,
                              hipStream_t stream) {
    const float* x  = (const float*)d_in[0];
    const int*   ei = (const int*)d_in[1];
    const float* W1 = (const float*)d_in[2]; const float* b1 = (const float*)d_in[3];
    const float* W2 = (const float*)d_in[4]; const float* b2 = (const float*)d_in[5];
    const float* Wp = (const float*)d_in[6]; const float* bp = (const float*)d_in[7];
    const float* Wc = (const float*)d_in[8]; const float* bc = (const float*)d_in[9];

    const int N = in_sizes[0] / 256;
    const int E = in_sizes[1] / 2;
    const int* srcE = ei;
    const int* dstE = ei + E;

    float* ws    = (float*)d_ws;
    float* dinv  = ws;                         // N
    float* buf1  = ws + N;                     // N x 64  (GEMM outputs H)
    float* buf2  = buf1 + (size_t)N * COLS;    // N x 64  (aggregation / next input)
    float* logit = (float*)d_out;              // N x 6
    float* zbuf  = (float*)d_out + (size_t)N * 6;  // N x 64 (second tuple element)

    const int B = 256;
    const int gN   = (N + B - 1) / B;
    const int gE   = (E + B - 1) / B;
    const int gNC  = (int)(((size_t)N * COLS + B - 1) / B);
    const int gEW  = (int)(((size_t)E * 32 + B - 1) / B);
    const int gGem = (N + 255) / 256;
    const size_t lds256 = 64 * (256 + 4) * sizeof(float);
    const size_t lds64  = 64 * (64 + 4)  * sizeof(float);

    // symmetric normalization
    k_fill1 <<<gN, B, 0, stream>>>(dinv, N);
    k_deg   <<<gE, B, 0, stream>>>(dstE, dinv, E);
    k_rsqrt <<<gN, B, 0, stream>>>(dinv, N);

    // layer 1: h = relu(agg(x @ W1) + b1)
    k_gemm_wmma<<<gGem, B, lds256, stream>>>(x, W1, nullptr, buf1, N, 256, 0);
    k_agg_init <<<gNC, B, 0, stream>>>(buf1, dinv, buf2, N);
    k_scatter  <<<gEW, B, 0, stream>>>(srcE, dstE, dinv, buf1, buf2, E);
    k_finalize <<<gNC, B, 0, stream>>>(buf2, b1, N);

    // layer 2
    k_gemm_wmma<<<gGem, B, lds64, stream>>>(buf2, W2, nullptr, buf1, N, 64, 0);
    k_agg_init <<<gNC, B, 0, stream>>>(buf1, dinv, buf2, N);
    k_scatter  <<<gEW, B, 0, stream>>>(srcE, dstE, dinv, buf1, buf2, E);
    k_finalize <<<gNC, B, 0, stream>>>(buf2, b2, N);

    // projection: z = relu(h @ Wp + bp)   (written directly to output tuple slot 1)
    k_gemm_wmma<<<gGem, B, lds64, stream>>>(buf2, Wp, bp, zbuf, N, 64, 1);

    // logits = z @ Wc + bc                (output tuple slot 0)
    k_logits<<<gN, B, 0, stream>>>(zbuf, Wc, bc, logit, N);
}